// TopLoss_283467842379
// MI455X (gfx1250) — compile-verified
//
#include <hip/hip_runtime.h>
#include <math.h>

typedef __attribute__((ext_vector_type(2))) float v2f;
typedef __attribute__((ext_vector_type(4))) float v4f;
typedef __attribute__((ext_vector_type(8))) float v8f;

#define BLOCKS_PER_ARRAY 1024
#define THREADS 256

// ---------------------------------------------------------------------------
// Kernel 1: streaming sum-of-squares (+ max for array 0) over barcode lengths.
// Memory-bound: 192 MB @ 23.3 TB/s ~ 8.2 us floor. Non-temporal b128 loads,
// register FMA accumulation, LDS -> WMMA(16x16x4 f32, A=ones) -> shfl reduce.
// ---------------------------------------------------------------------------
__global__ __launch_bounds__(THREADS)
void TopLoss_reduce_kernel(const float* __restrict__ d0,
                           const float* __restrict__ d1,
                           const float* __restrict__ d2,
                           int n0, int n1, int n2,   // pair counts per array
                           float* __restrict__ ws)
{
    const int ay = blockIdx.y;
    const float* src = (ay == 0) ? d0 : (ay == 1) ? d1 : d2;
    const int n_pairs = (ay == 0) ? n0 : (ay == 1) ? n1 : n2;

    const v4f* __restrict__ src4 = reinterpret_cast<const v4f*>(src);
    const int n4 = n_pairs >> 1;              // one v4f = 2 (birth,death) pairs
    const int tid = threadIdx.x;
    const int idx = blockIdx.x * THREADS + tid;
    const int stride = gridDim.x * THREADS;

    float acc = 0.0f;
    float mx  = -INFINITY;

    for (int i = idx; i < n4; i += stride) {
        v4f v = __builtin_nontemporal_load(&src4[i]);   // TH=NT: streamed once
        float la = v.x - v.y;
        float lb = v.z - v.w;
        la = isfinite(la) ? la : 0.0f;
        lb = isfinite(lb) ? lb : 0.0f;
        acc = fmaf(la, la, acc);
        acc = fmaf(lb, lb, acc);
        mx  = fmaxf(mx, fmaxf(la, lb));
    }
    // tail pair if pair count is odd (not the case for 8M, but stay correct)
    if (idx == 0 && (n_pairs & 1)) {
        float a = src[2 * (n_pairs - 1)];
        float b = src[2 * (n_pairs - 1) + 1];
        float l = a - b;
        l = isfinite(l) ? l : 0.0f;
        acc = fmaf(l, l, acc);
        mx  = fmaxf(mx, l);
    }

    __shared__ float ssum[THREADS];
    __shared__ float smax[THREADS];
    ssum[tid] = acc;
    smax[tid] = mx;
    __syncthreads();

    if (tid < 32) {   // wave 0 only: full EXEC inside (WMMA requirement)
        // 256 partials -> 32 lanes x 8 each, split 4/4 across B's two VGPRs
        float b0 = 0.0f, b1 = 0.0f, m = -INFINITY;
        const int base = tid * 8;
        #pragma unroll
        for (int j = 0; j < 4; ++j) b0 += ssum[base + j];
        #pragma unroll
        for (int j = 4; j < 8; ++j) b1 += ssum[base + j];
        #pragma unroll
        for (int j = 0; j < 8; ++j) m = fmaxf(m, smax[base + j]);

        // D = A(ones,16x4) x B(4x16) : every D[m][n] = sum of B column n.
        // Column n of B lives in lanes {n, n+16} x {VGPR0, VGPR1}, so one
        // WMMA reduces all 64 partials to 16 replicated column sums.
        v2f A; A[0] = 1.0f; A[1] = 1.0f;
        v2f B; B[0] = b0;   B[1] = b1;
        v8f C = {};
        C = __builtin_amdgcn_wmma_f32_16x16x4_f32(
                /*neg_a=*/false, A, /*neg_b=*/false, B,
                /*c_mod=*/(short)0, C, /*reuse_a=*/false, /*reuse_b=*/false);

        float s = C[0];                 // lane L holds column sum s_{L%16}
        s += __shfl_xor(s, 1, 32);      // sum the 16 distinct columns
        s += __shfl_xor(s, 2, 32);
        s += __shfl_xor(s, 4, 32);
        s += __shfl_xor(s, 8, 32);      // (mask 16 would double-count mirror)

        m = fmaxf(m, __shfl_xor(m, 1, 32));
        m = fmaxf(m, __shfl_xor(m, 2, 32));
        m = fmaxf(m, __shfl_xor(m, 4, 32));
        m = fmaxf(m, __shfl_xor(m, 8, 32));
        m = fmaxf(m, __shfl_xor(m, 16, 32));

        if (tid == 0) {
            ws[ay * gridDim.x + blockIdx.x] = s;                 // per-block SS
            if (ay == 0) ws[3 * gridDim.x + blockIdx.x] = m;     // per-block max
        }
    }
}

// ---------------------------------------------------------------------------
// Kernel 2: fold 1024 per-block partials per array + block maxima -> 5 outputs
// ---------------------------------------------------------------------------
__global__ __launch_bounds__(1024)
void TopLoss_finalize_kernel(const float* __restrict__ ws, int nblocks,
                             float* __restrict__ out)
{
    const int tid = threadIdx.x;
    float s0 = 0.0f, s1 = 0.0f, s2 = 0.0f, m = -INFINITY;
    for (int i = tid; i < nblocks; i += 1024) {
        s0 += ws[i];
        s1 += ws[nblocks + i];
        s2 += ws[2 * nblocks + i];
        m = fmaxf(m, ws[3 * nblocks + i]);
    }
    #pragma unroll
    for (int off = 1; off < 32; off <<= 1) {
        s0 += __shfl_xor(s0, off, 32);
        s1 += __shfl_xor(s1, off, 32);
        s2 += __shfl_xor(s2, off, 32);
        m = fmaxf(m, __shfl_xor(m, off, 32));
    }
    __shared__ float ls0[32], ls1[32], ls2[32], lm[32];
    const int wave = tid >> 5, lane = tid & 31;
    if (lane == 0) { ls0[wave] = s0; ls1[wave] = s1; ls2[wave] = s2; lm[wave] = m; }
    __syncthreads();
    if (tid < 32) {
        s0 = ls0[tid]; s1 = ls1[tid]; s2 = ls2[tid]; m = lm[tid];
        #pragma unroll
        for (int off = 1; off < 32; off <<= 1) {
            s0 += __shfl_xor(s0, off, 32);
            s1 += __shfl_xor(s1, off, 32);
            s2 += __shfl_xor(s2, off, 32);
            m = fmaxf(m, __shfl_xor(m, off, 32));
        }
        if (tid == 0) {
            const float m2  = m * m;
            const float t01 = 1.0f - m2;     // sum(1 - top1^2), top1 has 1 elem
            const float t0  = s0 - m2;       // sum(sortl0[1:]^2) = SS - max^2
            const float loss = t01 + t0 + s1 + s2;
            out[0] = loss;
            out[1] = t01;
            out[2] = t0;
            out[3] = s1;
            out[4] = s2;
        }
    }
}

extern "C" void kernel_launch(void* const* d_in, const int* in_sizes, int n_in,
                              void* d_out, int out_size, void* d_ws, size_t ws_size,
                              hipStream_t stream) {
    const float* d0 = (const float*)d_in[0];
    const float* d1 = (const float*)d_in[1];
    const float* d2 = (const float*)d_in[2];
    const int n0 = in_sizes[0] / 2;   // (N,2) -> N pairs
    const int n1 = in_sizes[1] / 2;
    const int n2 = in_sizes[2] / 2;
    float* ws  = (float*)d_ws;        // [3*B sums][B maxima], fully rewritten each call
    float* out = (float*)d_out;

    dim3 grid(BLOCKS_PER_ARRAY, 3, 1);
    TopLoss_reduce_kernel<<<grid, THREADS, 0, stream>>>(d0, d1, d2, n0, n1, n2, ws);
    TopLoss_finalize_kernel<<<1, 1024, 0, stream>>>(ws, BLOCKS_PER_ARRAY, out);
}